// BiLSTM_73787538145447
// MI455X (gfx1250) — compile-verified
//
#include <hip/hip_runtime.h>
#include <math.h>

// Problem constants (from reference): T=512, B=64, I=1024, H=1024
#define TT 512
#define BB 64
#define II 1024
#define HH 1024
#define G4 4096                 // 4*H gate width
#define BH (BB*HH)              // 65536
#define KSLABS 256              // II/4 == HH/4 k-slabs of 4
#define WMAT ((size_t)G4*II)    // floats per weight matrix (16 MB)

typedef float v2f __attribute__((ext_vector_type(2)));
typedef float v8f __attribute__((ext_vector_type(8)));

static __device__ __forceinline__ v2f ld2(const float* p) {
    return *reinterpret_cast<const v2f*>(p);   // 8B-aligned by construction
}

// D = A(16x4 f32) * B(4x16 f32) + C(16x16 f32)   -> v_wmma_f32_16x16x4_f32
static __device__ __forceinline__ v8f wmma4(v2f a, v2f b, v8f c) {
    return __builtin_amdgcn_wmma_f32_16x16x4_f32(
        false, a, false, b, (short)0, c, false, false);
}

// ---- packed-fragment index helpers -----------------------------------------
// B-fragment pack (weights): tile rt (16 rows), k-slab kb, lane L, sub s:
//   value = M[rt*16 + (L&15)][kb*4 + ((L>>4)<<1) + s]
static __device__ __forceinline__ size_t bpk(int rt, int kb) {
    return ((size_t)rt * KSLABS + kb) * 64;         // + lane*2 + s
}
// A-fragment pack (h state): element (row b, col n) of the h matrix:
static __device__ __forceinline__ size_t apk_elem(int b, int n) {
    int mt = b >> 4, kb = n >> 2;
    int L  = (b & 15) + (((n >> 1) & 1) << 4);
    int s  = n & 1;
    return (((size_t)mt * KSLABS + kb) * 32 + L) * 2 + s;
}

// ---------------------------------------------------------------------------
// Repack a (4096 x 1024) row-major weight matrix into WMMA B-fragment order.
// dst index == tid:  s = tid&1, L = (tid>>1)&31, kb = (tid>>6)&255, rt = tid>>14
__global__ void pack_w(const float* __restrict__ src, float* __restrict__ dst) {
    int tid = blockIdx.x * blockDim.x + threadIdx.x;    // 0 .. WMAT-1
    int s  = tid & 1;
    int L  = (tid >> 1) & 31;
    int kb = (tid >> 6) & 255;
    int rt = tid >> 14;
    int row = rt * 16 + (L & 15);
    int k   = kb * 4 + ((L >> 4) << 1) + s;
    dst[tid] = src[(size_t)row * II + k];
}

// ---------------------------------------------------------------------------
// Pack a chunk of x into A-fragment lane order.
// XP[slot][mt][kb][L][s],  slot = sl*2+dir,  t = dir ? T-1-(s0+sl) : s0+sl
// dst index == tid: s=tid&1, L=(tid>>1)&31, kb=(tid>>6)&255, mt=(tid>>14)&3, slot=tid>>16
__global__ void pack_x(const float* __restrict__ x, float* __restrict__ XP, int s0) {
    int tid = blockIdx.x * blockDim.x + threadIdx.x;    // 0 .. Tc*2*BB*II-1
    int s    = tid & 1;
    int L    = (tid >> 1) & 31;
    int kb   = (tid >> 6) & 255;
    int mt   = (tid >> 14) & 3;
    int slot = tid >> 16;
    int dir = slot & 1, sl = slot >> 1;
    int st = s0 + sl;
    int t = dir ? (TT - 1 - st) : st;
    int b = mt * 16 + (L & 15);
    int k = kb * 4 + ((L >> 4) << 1) + s;
    XP[tid] = x[((size_t)t * BB + b) * II + k];
}

// ---------------------------------------------------------------------------
// Initial state: hn -> hbuf (parity 0, packed A-layout if PK), cn -> cbuf.
template<bool PK>
__global__ void lstm_init(const float* __restrict__ hn, const float* __restrict__ cn,
                          float* __restrict__ hbuf, float* __restrict__ cbuf) {
    int i = blockIdx.x * blockDim.x + threadIdx.x;      // 0 .. 2*BH-1
    int dir = i / BH, idx = i % BH;
    int b = idx / HH, n = idx % HH;
    size_t off = PK ? apk_elem(b, n) : (size_t)idx;
    hbuf[(size_t)(dir * 2) * BH + off] = hn[i];
    cbuf[i] = cn[i];
}

// ---------------------------------------------------------------------------
// Packed pregate GEMM: one wave owns (column tile nt, ALL 4 batch tiles) of
// one (step,dir) slot: per k-slab it loads 1 B fragment + 4 A fragments and
// issues 4 WMMAs (1.25 loads/WMMA, all coalesced).
// grid = (G4/16, chunkSteps*2), block = 32
__global__ void lstm_pregate_pk(const float* __restrict__ XP,
                                const float* __restrict__ bWf, const float* __restrict__ bRf,
                                const float* __restrict__ bWb, const float* __restrict__ bRb,
                                const float* __restrict__ WpF, const float* __restrict__ WpB,
                                float* __restrict__ P) {
    int nt   = blockIdx.x;             // row tile of 4096 (0..255)
    int slot = blockIdx.y;             // sl*2+dir
    int dir  = slot & 1;
    int lane = threadIdx.x;
    int half = lane >> 4, lm = lane & 15;

    const float* wp = (dir ? WpB : WpF) + bpk(nt, 0) + lane * 2;
    const float* xp = XP + ((size_t)slot * 4) * KSLABS * 64 + lane * 2;

    v8f a0 = {0.f,0.f,0.f,0.f,0.f,0.f,0.f,0.f};
    v8f a1 = a0, a2 = a0, a3 = a0;
#pragma unroll 4
    for (int kb = 0; kb < KSLABS; ++kb) {
        v2f bv = ld2(wp + (size_t)kb * 64);
        a0 = wmma4(ld2(xp + ((size_t)(0 * KSLABS) + kb) * 64), bv, a0);
        a1 = wmma4(ld2(xp + ((size_t)(1 * KSLABS) + kb) * 64), bv, a1);
        a2 = wmma4(ld2(xp + ((size_t)(2 * KSLABS) + kb) * 64), bv, a2);
        a3 = wmma4(ld2(xp + ((size_t)(3 * KSLABS) + kb) * 64), bv, a3);
    }

    int n = nt * 16 + lm;
    float bias = dir ? (bWb[n] + bRb[n]) : (bWf[n] + bRf[n]);
    float* prow = P + (size_t)slot * BB * G4 + n;
#pragma unroll
    for (int mt = 0; mt < 4; ++mt) {
        v8f acc = mt == 0 ? a0 : (mt == 1 ? a1 : (mt == 2 ? a2 : a3));
        int mbase = mt * 16 + half * 8;
#pragma unroll
        for (int r = 0; r < 8; ++r)
            prow[(size_t)(mbase + r) * G4] = acc[r] + bias;
    }
}

// Direct (unpacked) fallback pregate: grid = (G4/16, BB/16, chunkSteps*2)
__global__ void lstm_pregate_direct(const float* __restrict__ x,
                                    const float* __restrict__ Wf, const float* __restrict__ bWf,
                                    const float* __restrict__ bRf,
                                    const float* __restrict__ Wb, const float* __restrict__ bWb,
                                    const float* __restrict__ bRb,
                                    float* __restrict__ P, int s0) {
    int nt = blockIdx.x;
    int mt = blockIdx.y;
    int z  = blockIdx.z;
    int dir = z & 1, sl = z >> 1;
    int s = s0 + sl;
    int t = dir ? (TT - 1 - s) : s;

    int lane = threadIdx.x;
    int half = lane >> 4, lm = lane & 15;

    const float* xrow = x + ((size_t)t * BB + (mt * 16 + lm)) * II + half * 2;
    const float* wrow = (dir ? Wb : Wf) + (size_t)(nt * 16 + lm) * II + half * 2;

    v8f a0 = {0.f,0.f,0.f,0.f,0.f,0.f,0.f,0.f};
    v8f a1 = a0, a2 = a0, a3 = a0;
    for (int kb = 0; kb < KSLABS; kb += 4) {
        a0 = wmma4(ld2(xrow + kb*4     ), ld2(wrow + kb*4     ), a0);
        a1 = wmma4(ld2(xrow + kb*4 + 4 ), ld2(wrow + kb*4 + 4 ), a1);
        a2 = wmma4(ld2(xrow + kb*4 + 8 ), ld2(wrow + kb*4 + 8 ), a2);
        a3 = wmma4(ld2(xrow + kb*4 + 12), ld2(wrow + kb*4 + 12), a3);
    }
    v8f acc = (a0 + a1) + (a2 + a3);

    int n = nt * 16 + lm;
    float bias = dir ? (bWb[n] + bRb[n]) : (bWf[n] + bRf[n]);
    float* prow = P + (size_t)(sl * 2 + dir) * BB * G4 + n;
    int mbase = mt * 16 + half * 8;
#pragma unroll
    for (int r = 0; r < 8; ++r)
        prow[(size_t)(mbase + r) * G4] = acc[r] + bias;
}

// ---------------------------------------------------------------------------
// One recurrence step, fused, split-K x4:
//   block = 128 (4 waves). Wave w computes the K-range [w*256, w*256+256)
//   of h@R.T for all 4 gates of one 16x16 tile; partials reduced via LDS,
//   then all 128 threads apply the LSTM cell update.
// grid = (HH/16, BB/16, 2 dirs)
template<bool PK>
__global__ void lstm_step(const float* __restrict__ P,
                          const float* __restrict__ Rf, const float* __restrict__ Rb,
                          const float* __restrict__ RpF, const float* __restrict__ RpB,
                          float* __restrict__ hbuf, float* __restrict__ cbuf,
                          float* __restrict__ out, int sl, int s, int par) {
    __shared__ __align__(32) float red[4][4][32][8];   // [wave][gate][lane][r] = 16KB

    int nt  = blockIdx.x;              // hidden tile (0..63)
    int mt  = blockIdx.y;              // batch tile (0..3)
    int dir = blockIdx.z;
    int tid = threadIdx.x;
    int w = tid >> 5, lane = tid & 31;
    int half = lane >> 4, lm = lane & 15;

    const float* hprev = hbuf + (size_t)(dir * 2 + par) * BH;
    float*       hnext = hbuf + (size_t)(dir * 2 + (par ^ 1)) * BH;
    float*       c     = cbuf + (size_t)dir * BH;

    v8f ai = {0.f,0.f,0.f,0.f,0.f,0.f,0.f,0.f};
    v8f af = ai, ag = ai, ao = ai;

    if (PK) {
        const float* Rp = dir ? RpB : RpF;
        const float* ha = hprev + ((size_t)mt * KSLABS + w * 64) * 64 + lane * 2;
        const float* b0 = Rp + bpk(0 * 64 + nt, w * 64) + lane * 2;   // gate i
        const float* b1 = Rp + bpk(1 * 64 + nt, w * 64) + lane * 2;   // gate f
        const float* b2 = Rp + bpk(2 * 64 + nt, w * 64) + lane * 2;   // gate g
        const float* b3 = Rp + bpk(3 * 64 + nt, w * 64) + lane * 2;   // gate o
#pragma unroll 4
        for (int kb = 0; kb < 64; ++kb) {          // all loads fully coalesced
            v2f hv = ld2(ha + (size_t)kb * 64);
            ai = wmma4(hv, ld2(b0 + (size_t)kb * 64), ai);
            af = wmma4(hv, ld2(b1 + (size_t)kb * 64), af);
            ag = wmma4(hv, ld2(b2 + (size_t)kb * 64), ag);
            ao = wmma4(hv, ld2(b3 + (size_t)kb * 64), ao);
        }
    } else {
        int n = nt * 16 + lm;
        const float* R    = dir ? Rb : Rf;
        const float* hrow = hprev + (size_t)(mt * 16 + lm) * HH + half * 2 + w * 256;
        const float* r0 = R + (size_t)(0 * HH + n) * HH + half * 2 + w * 256;
        const float* r1 = R + (size_t)(1 * HH + n) * HH + half * 2 + w * 256;
        const float* r2 = R + (size_t)(2 * HH + n) * HH + half * 2 + w * 256;
        const float* r3 = R + (size_t)(3 * HH + n) * HH + half * 2 + w * 256;
#pragma unroll 4
        for (int k = 0; k < 256; k += 4) {
            v2f hv = ld2(hrow + k);
            ai = wmma4(hv, ld2(r0 + k), ai);
            af = wmma4(hv, ld2(r1 + k), af);
            ag = wmma4(hv, ld2(r2 + k), ag);
            ao = wmma4(hv, ld2(r3 + k), ao);
        }
    }

    *(v8f*)&red[w][0][lane][0] = ai;
    *(v8f*)&red[w][1][lane][0] = af;
    *(v8f*)&red[w][2][lane][0] = ag;
    *(v8f*)&red[w][3][lane][0] = ao;
    __syncthreads();

    const float* prow = P + (size_t)(sl * 2 + dir) * BB * G4;
    int t = dir ? (TT - 1 - s) : s;
#pragma unroll
    for (int e = tid; e < 256; e += 128) {         // 2 cell updates per thread
        int rb = e >> 4, ln = e & 15;              // local (row, col) in tile
        int lidx = ln + ((rb >> 3) << 4);
        int r = rb & 7;
        float iv = 0.f, fv = 0.f, gv = 0.f, ov = 0.f;
#pragma unroll
        for (int w2 = 0; w2 < 4; ++w2) {
            iv += red[w2][0][lidx][r];
            fv += red[w2][1][lidx][r];
            gv += red[w2][2][lidx][r];
            ov += red[w2][3][lidx][r];
        }
        int b = mt * 16 + rb, nn = nt * 16 + ln;
        const float* pg = prow + (size_t)b * G4;
        iv += pg[0 * HH + nn];
        fv += pg[1 * HH + nn];
        gv += pg[2 * HH + nn];
        ov += pg[3 * HH + nn];
        float co = c[(size_t)b * HH + nn];
        float si = 1.f / (1.f + expf(-iv));
        float sf = 1.f / (1.f + expf(-fv));
        float so = 1.f / (1.f + expf(-ov));
        float c2 = sf * co + si * tanhf(gv);
        float h2 = so * tanhf(c2);
        c[(size_t)b * HH + nn] = c2;
        if (PK) hnext[apk_elem(b, nn)] = h2;       // next step's A-fragment order
        else    hnext[(size_t)b * HH + nn] = h2;
        out[((size_t)t * BB + b) * (2 * HH) + (size_t)dir * HH + nn] = h2;
    }
}

// ---------------------------------------------------------------------------
// hidden_forward = [hf; cf], hidden_backward = [hb; cb] -> d_out tail.
template<bool PK>
__global__ void lstm_final(const float* __restrict__ hbuf, const float* __restrict__ cbuf,
                           float* __restrict__ tail, int par) {
    int i = blockIdx.x * blockDim.x + threadIdx.x;      // 0 .. 4*BH-1
    int dir = i / (2 * BH);
    int rem = i % (2 * BH);
    int hc  = rem / BH;
    int idx = rem % BH;
    float v;
    if (hc == 0) {
        int b = idx / HH, n = idx % HH;
        size_t off = PK ? apk_elem(b, n) : (size_t)idx;
        v = hbuf[(size_t)(dir * 2 + par) * BH + off];
    } else {
        v = cbuf[(size_t)dir * BH + idx];
    }
    tail[i] = v;
}

// ---------------------------------------------------------------------------
extern "C" void kernel_launch(void* const* d_in, const int* in_sizes, int n_in,
                              void* d_out, int out_size, void* d_ws, size_t ws_size,
                              hipStream_t stream) {
    (void)in_sizes; (void)n_in; (void)out_size;
    const float* x    = (const float*)d_in[0];
    const float* hn   = (const float*)d_in[1];
    const float* cn   = (const float*)d_in[2];
    const float* W_f  = (const float*)d_in[3];
    const float* R_f  = (const float*)d_in[4];
    const float* Wb_f = (const float*)d_in[5];
    const float* Rb_f = (const float*)d_in[6];
    const float* W_b  = (const float*)d_in[7];
    const float* R_b  = (const float*)d_in[8];
    const float* Wb_b = (const float*)d_in[9];
    const float* Rb_b = (const float*)d_in[10];
    float* out = (float*)d_out;
    float* ws  = (float*)d_ws;

    // ws layout: hbuf | cbuf | (packed W/R) | P chunk | XP chunk
    float* hbuf = ws;
    float* cbuf = ws + (size_t)4 * BH;
    size_t baseFloats = (size_t)6 * BH;
    size_t packFloats = baseFloats + 4 * WMAT;          // + 64MB of packed weights

    const size_t stepP  = (size_t)2 * BB * G4;          // P floats per chunk step
    const size_t stepXP = (size_t)2 * BB * II;          // XP floats per chunk step

    bool pk = ws_size >= (packFloats + stepP + stepXP) * sizeof(float);
    float *WpF = nullptr, *WpB = nullptr, *RpF = nullptr, *RpB = nullptr;
    float *P, *XP = nullptr;
    size_t fixedFloats = pk ? packFloats : baseFloats;
    if (pk) {
        WpF = ws + baseFloats;
        WpB = WpF + WMAT;
        RpF = WpB + WMAT;
        RpB = RpF + WMAT;
    }
    P = ws + fixedFloats;

    // Largest power-of-two chunk that fits; cap at 32 steps so pregate chunk
    // (+ packed x slice) + 64 MB of weights stay resident in the 192 MB L2.
    int Tc = 1;
    size_t perStep = pk ? (stepP + stepXP) : stepP;
    for (int cch = 32; cch >= 1; cch >>= 1) {
        size_t need = (fixedFloats + (size_t)cch * perStep) * sizeof(float);
        if (need <= ws_size) { Tc = cch; break; }
    }
    if (pk) XP = P + (size_t)Tc * stepP;

    if (pk) {
        int nb = (int)(WMAT / 256);
        pack_w<<<nb, 256, 0, stream>>>(W_f, WpF);
        pack_w<<<nb, 256, 0, stream>>>(W_b, WpB);
        pack_w<<<nb, 256, 0, stream>>>(R_f, RpF);
        pack_w<<<nb, 256, 0, stream>>>(R_b, RpB);
        lstm_init<true><<<(2 * BH) / 256, 256, 0, stream>>>(hn, cn, hbuf, cbuf);
    } else {
        lstm_init<false><<<(2 * BH) / 256, 256, 0, stream>>>(hn, cn, hbuf, cbuf);
    }

    int par = 0;
    for (int s0 = 0; s0 < TT; s0 += Tc) {
        if (pk) {
            int nelt = (int)((size_t)Tc * stepXP);
            pack_x<<<nelt / 256, 256, 0, stream>>>(x, XP, s0);
            lstm_pregate_pk<<<dim3(G4 / 16, Tc * 2), 32, 0, stream>>>(
                XP, Wb_f, Rb_f, Wb_b, Rb_b, WpF, WpB, P);
        } else {
            lstm_pregate_direct<<<dim3(G4 / 16, BB / 16, Tc * 2), 32, 0, stream>>>(
                x, W_f, Wb_f, Rb_f, W_b, Wb_b, Rb_b, P, s0);
        }
        for (int sl = 0; sl < Tc; ++sl) {
            if (pk)
                lstm_step<true><<<dim3(HH / 16, BB / 16, 2), 128, 0, stream>>>(
                    P, R_f, R_b, RpF, RpB, hbuf, cbuf, out, sl, s0 + sl, par);
            else
                lstm_step<false><<<dim3(HH / 16, BB / 16, 2), 128, 0, stream>>>(
                    P, R_f, R_b, RpF, RpB, hbuf, cbuf, out, sl, s0 + sl, par);
            par ^= 1;
        }
    }

    if (pk)
        lstm_final<true><<<(4 * BH) / 256, 256, 0, stream>>>(
            hbuf, cbuf, out + (size_t)TT * BB * 2 * HH, par);
    else
        lstm_final<false><<<(4 * BH) / 256, 256, 0, stream>>>(
            hbuf, cbuf, out + (size_t)TT * BB * 2 * HH, par);
}